// Clopath_71717363909258
// MI455X (gfx1250) — compile-verified
//
#include <hip/hip_runtime.h>
#include <hip/hip_bf16.h>

typedef float v2f __attribute__((ext_vector_type(2)));
typedef float v8f __attribute__((ext_vector_type(8)));

#define D_    8
#define B_    32
#define PRE_  1024
#define POST_ 1024
#define EO_   (PRE_ * POST_)   // 1048576
#define BP_   (B_ * PRE_)      // 32768

// exp(-1/10), exp(-1/20), exp(-1/15) and complements
#define ALPHA_P 0.9048374180f
#define ONEM_P  0.0951625820f
#define ALPHA_D 0.9512294245f
#define ONEM_D  0.0487705755f
#define ALPHA_X 0.9355069850f
#define ONEM_X  0.0644930150f

__device__ __forceinline__ v8f wmma_f32_16x16x4(v2f a, v2f b, v8f c) {
    // (neg_a, A, neg_b, B, c_mod, C, reuse_a, reuse_b)
    return __builtin_amdgcn_wmma_f32_16x16x4_f32(false, a, false, b, (short)0, c,
                                                 false, false);
}

// --- small kernel 1: gates + u_pot/u_dep filters (B*POST elements) ---------
__global__ __launch_bounds__(256) void gates_u_kernel(
    const float* __restrict__ Xpost, const float* __restrict__ Vpost,
    const float* __restrict__ u_pot, const float* __restrict__ u_dep,
    float* __restrict__ gp, float* __restrict__ gd,
    float* __restrict__ out_upot, float* __restrict__ out_udep)
{
    int i = blockIdx.x * blockDim.x + threadIdx.x;
    if (i < B_ * POST_) {
        float up = u_pot[i], ud = u_dep[i];
        float xp = Xpost[i], vp = Vpost[i];
        gp[i] = xp * fmaxf(up, 0.0f);
        gd[i] = fmaxf(ud, 0.0f);
        out_upot[i] = ALPHA_P * up + ONEM_P * vp;
        out_udep[i] = ALPHA_D * ud + ONEM_D * vp;
    }
}

// --- small kernel 2: xbar_pre filter (D*B*PRE elements) --------------------
__global__ __launch_bounds__(256) void xbar_kernel(
    const float* __restrict__ xbar, const float* __restrict__ Xd,
    float* __restrict__ out_xbar)
{
    int i = blockIdx.x * blockDim.x + threadIdx.x;
    if (i < D_ * B_ * PRE_) {
        out_xbar[i] = ALPHA_X * xbar[i] + ONEM_X * Xd[i];
    }
}

// --- main kernel: per-e 32x8 @ 8x16 WMMA contraction + W update ------------
// One wave handles one 16-wide o tile and 32 consecutive e, all 32 b rows.
// 2048 waves = 64 o-tiles x 32 e-chunks. 256 threads/block -> 256 blocks.
__global__ __launch_bounds__(256) void clopath_main_kernel(
    const float* __restrict__ Xd,   const float* __restrict__ W,
    const float* __restrict__ xbar, const float* __restrict__ dmap,
    const float* __restrict__ Ap,   const float* __restrict__ Ad,
    const float* __restrict__ wmax, const float* __restrict__ gp,
    const float* __restrict__ gd,
    float* __restrict__ out_w, float* __restrict__ out_wnew)
{
    const int lane = threadIdx.x & 31;
    const int wave = blockIdx.x * (blockDim.x >> 5) + (threadIdx.x >> 5);
    if (wave >= 64 * 32) return;               // uniform per-wave guard

    const int t_o     = wave & 63;             // o16 tile index
    const int e_chunk = wave >> 6;             // 32 e per chunk
    const int o0 = t_o << 4;
    const int e0 = e_chunk << 5;

    const int lo = lane & 15;                  // N / M within half-wave
    const int hi = lane >> 4;                  // half-wave select
    const int o_lane = o0 + lo;

    // Hoist gates: C/D element r of M-tile m maps to b = m*16 + hi*8 + r.
    float gpv[2][8], gdv[2][8];
    #pragma unroll
    for (int m = 0; m < 2; ++m) {
        #pragma unroll
        for (int r = 0; r < 8; ++r) {
            int b = m * 16 + hi * 8 + r;
            gpv[m][r] = gp[b * POST_ + o_lane];
            gdv[m][r] = gd[b * POST_ + o_lane];
        }
    }

    const v8f zero = {0.f, 0.f, 0.f, 0.f, 0.f, 0.f, 0.f, 0.f};

    for (int i = 0; i < 32; ++i) {
        const int  e       = e0 + i;
        const long eo_base = (long)e * POST_ + o_lane;

        // B operands (dmap[d,e,o], KxN): lane<16 -> K={0,1}(+4), lane>=16 -> K={2,3}(+4)
        v2f b0, b1;
        b0[0] = dmap[(long)(hi * 2 + 0) * EO_ + eo_base];
        b0[1] = dmap[(long)(hi * 2 + 1) * EO_ + eo_base];
        b1[0] = dmap[(long)(hi * 2 + 4) * EO_ + eo_base];
        b1[1] = dmap[(long)(hi * 2 + 5) * EO_ + eo_base];

        v8f spot[2], sdep[2];
        #pragma unroll
        for (int m = 0; m < 2; ++m) {
            const int  brow   = m * 16 + lo;            // A row M = b
            const long a_base = (long)brow * PRE_ + e;  // xbar/Xd: [d, b, e]
            v2f ax0, ax1, av0, av1;
            ax0[0] = xbar[(long)(hi * 2 + 0) * BP_ + a_base];
            ax0[1] = xbar[(long)(hi * 2 + 1) * BP_ + a_base];
            ax1[0] = xbar[(long)(hi * 2 + 4) * BP_ + a_base];
            ax1[1] = xbar[(long)(hi * 2 + 5) * BP_ + a_base];
            av0[0] = Xd  [(long)(hi * 2 + 0) * BP_ + a_base];
            av0[1] = Xd  [(long)(hi * 2 + 1) * BP_ + a_base];
            av1[0] = Xd  [(long)(hi * 2 + 4) * BP_ + a_base];
            av1[1] = Xd  [(long)(hi * 2 + 5) * BP_ + a_base];

            v8f cp = wmma_f32_16x16x4(ax0, b0, zero);   // K = 0..3
            cp     = wmma_f32_16x16x4(ax1, b1, cp);     // K = 4..7
            v8f cd = wmma_f32_16x16x4(av0, b0, zero);
            cd     = wmma_f32_16x16x4(av1, b1, cd);
            spot[m] = cp;
            sdep[m] = cd;
        }

        const float ap = Ap[eo_base];
        const float ad = Ad[eo_base];
        const float wm = wmax[eo_base];

        #pragma unroll
        for (int m = 0; m < 2; ++m) {
            #pragma unroll
            for (int r = 0; r < 8; ++r) {
                const int  b  = m * 16 + hi * 8 + r;
                const long wi = (long)b * EO_ + eo_base;
                const float w = W[wi];
                out_w[wi] = w;                          // out = W (pre-update)
                float dw = spot[m][r] * ap * gpv[m][r]
                         - sdep[m][r] * ad * gdv[m][r];
                float wn = fminf(wm, fmaxf(w + dw, 0.0f));
                out_wnew[wi] = wn;
            }
        }
    }
}

extern "C" void kernel_launch(void* const* d_in, const int* in_sizes, int n_in,
                              void* d_out, int out_size, void* d_ws, size_t ws_size,
                              hipStream_t stream) {
    (void)in_sizes; (void)n_in; (void)out_size; (void)ws_size;
    const float* Xd    = (const float*)d_in[0];
    const float* Xpost = (const float*)d_in[1];
    const float* Vpost = (const float*)d_in[2];
    const float* W     = (const float*)d_in[3];
    const float* xbar  = (const float*)d_in[4];
    const float* u_pot = (const float*)d_in[5];
    const float* u_dep = (const float*)d_in[6];
    const float* dmap  = (const float*)d_in[7];
    const float* Ap    = (const float*)d_in[8];
    const float* Ad    = (const float*)d_in[9];
    const float* wmax  = (const float*)d_in[10];

    float* out      = (float*)d_out;
    float* out_w    = out;                                   // B*PRE*POST
    float* out_wnew = out + (long)B_ * EO_;                  // B*PRE*POST
    float* out_xbar = out + 2L * B_ * EO_;                   // D*B*PRE
    float* out_upot = out_xbar + (long)D_ * BP_;             // B*POST
    float* out_udep = out_upot + (long)B_ * POST_;           // B*POST

    float* gp = (float*)d_ws;                                // B*POST
    float* gd = gp + B_ * POST_;                             // B*POST

    gates_u_kernel<<<(B_ * POST_ + 255) / 256, 256, 0, stream>>>(
        Xpost, Vpost, u_pot, u_dep, gp, gd, out_upot, out_udep);

    xbar_kernel<<<(D_ * B_ * PRE_ + 255) / 256, 256, 0, stream>>>(
        xbar, Xd, out_xbar);

    // 2048 waves: 256 blocks x 8 waves
    clopath_main_kernel<<<256, 256, 0, stream>>>(
        Xd, W, xbar, dmap, Ap, Ad, wmax, gp, gd, out_w, out_wnew);
}